// GNNmodel_16174846836757
// MI455X (gfx1250) — compile-verified
//
#include <hip/hip_runtime.h>

typedef __attribute__((ext_vector_type(2))) float v2f;
typedef __attribute__((ext_vector_type(4))) float v4f;
typedef __attribute__((ext_vector_type(8))) float v8f;

#define GSLOTS 8

// ---------------------------------------------------------------- utility
__global__ void k_fill(float* __restrict__ p, float v, int n) {
    int i = blockIdx.x * blockDim.x + threadIdx.x;
    if (i < n) p[i] = v;
}

// deg[dst] += 1 for every edge (self-loop "1" pre-filled by k_fill).
// Index stream is read-once: non-temporal so it doesn't evict reused data from L2.
__global__ void k_deg_edges(const long long* __restrict__ dst, float* __restrict__ deg, int ne) {
    int e = blockIdx.x * blockDim.x + threadIdx.x;
    if (e < ne) {
        long long d = __builtin_nontemporal_load(dst + e);
        atomicAdd(&deg[d], 1.0f);
    }
}

// in-place deg -> deg^{-1/2}  (deg >= 1 always: self-loops)
__global__ void k_rsqrt(float* __restrict__ d, int n) {
    int i = blockIdx.x * blockDim.x + threadIdx.x;
    if (i < n) d[i] = rsqrtf(d[i]);
}

// ---------------------------------------------------------------- layer-1 GEMM
// t1[N,16] = x[N,2] @ W1[2,16]    one 16x16 tile per wave, K=2 zero-padded to 4.
// Only full 16-row tiles reach this kernel -> no store guards (no exec diamonds).
__global__ void k_gemm1(const float* __restrict__ x, const float* __restrict__ W1,
                        float* __restrict__ t1, int ntiles) {
    int wave = (int)((blockIdx.x * blockDim.x + threadIdx.x) >> 5);
    if (wave >= ntiles) return;                 // wave-uniform exit
    int lane = threadIdx.x & 31;
    int col  = lane & 15;
    bool hi  = lane >= 16;
    int row  = wave * 16 + col;

    // A 16x4 (f32): lanes0-15 -> K=0,1 ; lanes16-31 -> K=2,3 (zero pad)
    // B 4x16: VGPR0 rows K=0(lo)/K=2(hi), VGPR1 rows K=1(lo)/K=3(hi)
    v2f a; a.x = 0.0f; a.y = 0.0f;
    v2f b; b.x = 0.0f; b.y = 0.0f;
    if (!hi) {
        a = ((const v2f*)x)[row];               // global_load_b64
        b.x = W1[col]; b.y = W1[16 + col];
    }

    v8f c = {};
    c = __builtin_amdgcn_wmma_f32_16x16x4_f32(false, a, false, b, (short)0, c, false, false);

    int mbase = wave * 16 + (hi ? 8 : 0);
#pragma unroll
    for (int j = 0; j < 8; ++j) t1[(mbase + j) * 16 + col] = c[j];
}

// scalar tail for the (n % 16) leftover rows (not launched when n % 16 == 0)
__global__ void k_tail1(const float* __restrict__ x, const float* __restrict__ W1,
                        float* __restrict__ t1, int start, int n) {
    int idx = blockIdx.x * blockDim.x + threadIdx.x;
    int r = start + (idx >> 4), c = idx & 15;
    if (r >= n) return;
    t1[r * 16 + c] = x[r * 2 + 0] * W1[c] + x[r * 2 + 1] * W1[16 + c];
}

// ---------------------------------------------------------------- layer-2 GEMM
// t2[N,32] = relu(h1raw + b1) @ W2[16,32]   (bias+ReLU fused into A-load)
__global__ void k_gemm2(const float* __restrict__ h1raw, const float* __restrict__ b1,
                        const float* __restrict__ W2, float* __restrict__ t2, int ntiles) {
    int wave = (int)((blockIdx.x * blockDim.x + threadIdx.x) >> 5);
    if (wave >= ntiles) return;
    int lane = threadIdx.x & 31;
    int col  = lane & 15;
    bool hi  = lane >= 16;
    int row  = wave * 16 + col;

    v8f c0 = {}, c1 = {};
#pragma unroll
    for (int kk = 0; kk < 4; ++kk) {
        int kb = kk * 4 + (hi ? 2 : 0);
        v2f av = ((const v2f*)(h1raw + row * 16))[kb >> 1];   // global_load_b64, kb is even
        v2f a;
        a.x = fmaxf(av.x + b1[kb + 0], 0.0f);
        a.y = fmaxf(av.y + b1[kb + 1], 0.0f);
        v2f b0, b1v;
        b0.x  = W2[(kb + 0) * 32 + col];       b0.y  = W2[(kb + 1) * 32 + col];
        b1v.x = W2[(kb + 0) * 32 + 16 + col];  b1v.y = W2[(kb + 1) * 32 + 16 + col];
        c0 = __builtin_amdgcn_wmma_f32_16x16x4_f32(false, a, false, b0,  (short)0, c0, false, false);
        c1 = __builtin_amdgcn_wmma_f32_16x16x4_f32(false, a, false, b1v, (short)0, c1, false, false);
    }
    int mbase = wave * 16 + (hi ? 8 : 0);
#pragma unroll
    for (int j = 0; j < 8; ++j) {
        int r = mbase + j;
        t2[r * 32 + col]      = c0[j];
        t2[r * 32 + 16 + col] = c1[j];
    }
}

__global__ void k_tail2(const float* __restrict__ h1raw, const float* __restrict__ b1,
                        const float* __restrict__ W2, float* __restrict__ t2,
                        int start, int n) {
    int idx = blockIdx.x * blockDim.x + threadIdx.x;
    int r = start + (idx >> 5), c = idx & 31;
    if (r >= n) return;
    float s = 0.0f;
#pragma unroll
    for (int k = 0; k < 16; ++k)
        s += fmaxf(h1raw[r * 16 + k] + b1[k], 0.0f) * W2[k * 32 + c];
    t2[r * 32 + c] = s;
}

// ---------------------------------------------------------------- self-loop init
// out[n,:] = t[n,:] * dinv[n]^2
template <int F>
__global__ void k_self_init(const float* __restrict__ t, const float* __restrict__ dinv,
                            float* __restrict__ out, int n) {
    int i = blockIdx.x * blockDim.x + threadIdx.x;
    if (i >= n) return;
    float w = dinv[i]; w *= w;
    const float4* tr = (const float4*)(t + (size_t)i * F);
    float4* orow = (float4*)(out + (size_t)i * F);
#pragma unroll
    for (int j = 0; j < F / 4; ++j) {
        float4 v = tr[j];
        v.x *= w; v.y *= w; v.z *= w; v.w *= w;
        orow[j] = v;
    }
}

// ---------------------------------------------------------------- edge scatter
// out[dst,:] += t[src,:] * dinv[src]*dinv[dst]
// Index streams NT (read-once, 128MB); feature gathers stay RT so rows remain in L2.
template <int F>
__global__ void k_edge(const long long* __restrict__ src, const long long* __restrict__ dst,
                       const float* __restrict__ dinv, const float* __restrict__ t,
                       float* __restrict__ out, int ne) {
    int e = blockIdx.x * blockDim.x + threadIdx.x;
    if (e >= ne) return;
    long long s = __builtin_nontemporal_load(src + e);
    long long d = __builtin_nontemporal_load(dst + e);
    float w = dinv[s] * dinv[d];
    const float4* tr = (const float4*)(t + (size_t)s * F);
    float* orow = out + (size_t)d * F;
#pragma unroll
    for (int j = 0; j < F / 4; ++j) {
        float4 v = tr[j];
        atomicAdd(orow + 4 * j + 0, v.x * w);
        atomicAdd(orow + 4 * j + 1, v.y * w);
        atomicAdd(orow + 4 * j + 2, v.z * w);
        atomicAdd(orow + 4 * j + 3, v.w * w);
    }
}

// ---------------------------------------------------------------- pooling
// batch is sorted, so a 256-node block spans very few graphs: accumulate in LDS
// (ds_add_f32), flush the few used slots with global atomics.
__global__ void k_pool(const float* __restrict__ h2raw, const float* __restrict__ b2,
                       const long long* __restrict__ batch, float* __restrict__ pool,
                       float* __restrict__ cnt, int n) {
    __shared__ float sacc[GSLOTS * 32];
    __shared__ float scnt[GSLOTS];
    int tid  = threadIdx.x;
    int base = blockIdx.x * blockDim.x;
    for (int j = tid; j < GSLOTS * 32; j += blockDim.x) sacc[j] = 0.0f;
    if (tid < GSLOTS) scnt[tid] = 0.0f;
    __syncthreads();

    long long g0 = batch[base];
    int i = base + tid;
    if (i < n) {
        long long g = batch[i];
        long long off = g - g0;
        const v4f* hr = (const v4f*)(h2raw + (size_t)i * 32);
        if (off < GSLOTS) {
#pragma unroll
            for (int j = 0; j < 8; ++j) {
                v4f v = __builtin_nontemporal_load(hr + j);      // h2 is read-once
                atomicAdd(&sacc[(int)off * 32 + 4 * j + 0], fmaxf(v.x + b2[4 * j + 0], 0.0f));
                atomicAdd(&sacc[(int)off * 32 + 4 * j + 1], fmaxf(v.y + b2[4 * j + 1], 0.0f));
                atomicAdd(&sacc[(int)off * 32 + 4 * j + 2], fmaxf(v.z + b2[4 * j + 2], 0.0f));
                atomicAdd(&sacc[(int)off * 32 + 4 * j + 3], fmaxf(v.w + b2[4 * j + 3], 0.0f));
            }
            atomicAdd(&scnt[(int)off], 1.0f);
        } else {  // rare fallback
#pragma unroll
            for (int f = 0; f < 32; ++f)
                atomicAdd(&pool[g * 32 + f], fmaxf(h2raw[(size_t)i * 32 + f] + b2[f], 0.0f));
            atomicAdd(&cnt[g], 1.0f);
        }
    }
    __syncthreads();
    for (int j = tid; j < GSLOTS * 32; j += blockDim.x) {
        float v = sacc[j];
        if (v != 0.0f) atomicAdd(&pool[(g0 + (j >> 5)) * 32 + (j & 31)], v);
    }
    if (tid < GSLOTS) {
        float v = scnt[tid];
        if (v != 0.0f) atomicAdd(&cnt[g0 + tid], v);
    }
}

// ---------------------------------------------------------------- final linear
__global__ void k_final(const float* __restrict__ pool, const float* __restrict__ cnt,
                        const float* __restrict__ Wl, const float* __restrict__ bl,
                        float* __restrict__ out, int ng) {
    int g = blockIdx.x * blockDim.x + threadIdx.x;
    if (g >= ng) return;
    float c = fmaxf(cnt[g], 1.0f);
    float s = 0.0f;
#pragma unroll
    for (int f = 0; f < 32; ++f) s += pool[g * 32 + f] * Wl[f];
    out[g] = s / c + bl[0];
}

// ---------------------------------------------------------------- launcher
extern "C" void kernel_launch(void* const* d_in, const int* in_sizes, int n_in,
                              void* d_out, int out_size, void* d_ws, size_t ws_size,
                              hipStream_t stream) {
    const float*     x   = (const float*)d_in[0];
    const long long* ei  = (const long long*)d_in[1];
    const long long* bat = (const long long*)d_in[2];
    const float*     W1  = (const float*)d_in[3];
    const float*     b1  = (const float*)d_in[4];
    const float*     W2  = (const float*)d_in[5];
    const float*     b2  = (const float*)d_in[6];
    const float*     Wl  = (const float*)d_in[7];
    const float*     bl  = (const float*)d_in[8];
    float*           out = (float*)d_out;

    const int n  = in_sizes[0] / 2;        // 500,000 nodes
    const int ne = in_sizes[1] / 2;        // 8,000,000 edges
    const int ng = out_size;               // 1024 graphs
    const long long* src = ei;
    const long long* dst = ei + ne;

    // workspace carve-up (floats): dinv | t (N*32, reused t1->t2) | h (N*32, h1->h2) | pool | cnt
    float* ws   = (float*)d_ws;
    float* dinv = ws;
    float* t    = dinv + (size_t)n;
    float* h    = t + (size_t)n * 32;
    float* pool = h + (size_t)n * 32;
    float* cnt  = pool + (size_t)ng * 32;

    const int B = 256;
    int nbN = (n + B - 1) / B;
    int nbE = (ne + B - 1) / B;
    int ntiles = n / 16;                   // full WMMA tiles (N=500,000 -> 31,250 exact)
    int rem    = n - ntiles * 16;
    int nbT = (ntiles + 7) / 8;            // 8 waves per 256-thread block

    // degrees -> dinv
    k_fill<<<nbN, B, 0, stream>>>(dinv, 1.0f, n);                 // self-loop
    k_deg_edges<<<nbE, B, 0, stream>>>(dst, dinv, ne);
    k_rsqrt<<<nbN, B, 0, stream>>>(dinv, n);

    // layer 1: t1 = x@W1 ; h1 = selfloop + edge scatter
    k_gemm1<<<nbT, B, 0, stream>>>(x, W1, t, ntiles);
    if (rem) k_tail1<<<(rem * 16 + B - 1) / B, B, 0, stream>>>(x, W1, t, ntiles * 16, n);
    k_self_init<16><<<nbN, B, 0, stream>>>(t, dinv, h, n);
    k_edge<16><<<nbE, B, 0, stream>>>(src, dst, dinv, t, h, ne);

    // layer 2: t2 = relu(h1+b1)@W2 ; h2 = selfloop + edge scatter
    k_gemm2<<<nbT, B, 0, stream>>>(h, b1, W2, t, ntiles);
    if (rem) k_tail2<<<(rem * 32 + B - 1) / B, B, 0, stream>>>(h, b1, W2, t, ntiles * 16, n);
    k_self_init<32><<<nbN, B, 0, stream>>>(t, dinv, h, n);
    k_edge<32><<<nbE, B, 0, stream>>>(src, dst, dinv, t, h, ne);

    // pooling + final linear
    k_fill<<<(ng * 33 + B - 1) / B, B, 0, stream>>>(pool, 0.0f, ng * 33);  // pool + cnt contiguous
    k_pool<<<nbN, B, 0, stream>>>(h, b2, bat, pool, cnt, n);
    k_final<<<(ng + B - 1) / B, B, 0, stream>>>(pool, cnt, Wl, bl, out, ng);
}